// DoubleLayer_85822036508672
// MI455X (gfx1250) — compile-verified
//
#include <hip/hip_runtime.h>
#include <cstdint>
#include <cstddef>

// ---------------- problem constants ----------------
#define BS   8192
#define C    23
#define F    1024
#define EPS  1e-5f

// kernel C tiling
#define BTILE   128              // b-rows per block
#define FCHUNK  64               // f per staged chunk
#define XPAD    4                // LDS row padding (floats) -> conflict-free ds_load_b64
#define XROW    (FCHUNK + XPAD)  // 68
#define NCHUNK  (F / FCHUNK)     // 16

typedef float v2f __attribute__((ext_vector_type(2)));
typedef float v8f __attribute__((ext_vector_type(8)));
typedef int   v4i __attribute__((ext_vector_type(4)));

#define AS_GLOBAL __attribute__((address_space(1)))
#define AS_LDS    __attribute__((address_space(3)))

#ifndef __has_builtin
#define __has_builtin(x) 0
#endif

#if __has_builtin(__builtin_amdgcn_global_load_async_to_lds_b128)
#define HAVE_ASYNC 1
#else
#define HAVE_ASYNC 0
#endif

#if __has_builtin(__builtin_amdgcn_wmma_f32_16x16x4_f32)
#define HAVE_WMMA 1
#else
#define HAVE_WMMA 0
#endif

#if __has_builtin(__builtin_amdgcn_s_wait_asynccnt)
#define WAIT_ASYNC(n) __builtin_amdgcn_s_wait_asynccnt((unsigned short)(n))
#else
#define WAIT_ASYNC(n) asm volatile("s_wait_asynccnt %0" ::"i"(n) : "memory")
#endif

// workspace layout (in floats)
#define OFF_SUM    0
#define OFF_SUMSQ  (C * F)            // 23552
#define OFF_A      (2 * C * F)
#define OFF_T      (3 * C * F)        // 23 floats
#define OFF_U      (3 * C * F + 32)   // 23 floats
#define OFF_V      (3 * C * F + 64)   // 23 floats
#define OFF_H      (3 * C * F + 96)   // BS*C floats

// ---------------------------------------------------------------------------
// Kernel A: column sums over batch -> sum1[c,f], sumsq1[c,f]
// grid = C * (BS/128), block = 256; thread owns 4 consecutive f (float4 loads)
// ---------------------------------------------------------------------------
__global__ __launch_bounds__(256) void k_stats1(const float* __restrict__ x,
                                                float* __restrict__ sum1,
                                                float* __restrict__ sumsq1) {
  const int nb  = BS / 128;
  const int c   = blockIdx.x / nb;
  const int bch = blockIdx.x % nb;
  const int f   = threadIdx.x * 4;

  const float* p = x + ((size_t)(bch * 128) * C + c) * F + f;
  float s0 = 0.f, s1 = 0.f, s2 = 0.f, s3 = 0.f;
  float q0 = 0.f, q1 = 0.f, q2 = 0.f, q3 = 0.f;
  for (int i = 0; i < 128; ++i) {
    const float4 v = *(const float4*)p;
    p += (size_t)C * F;
    s0 += v.x; s1 += v.y; s2 += v.z; s3 += v.w;
    q0 += v.x * v.x; q1 += v.y * v.y; q2 += v.z * v.z; q3 += v.w * v.w;
  }
  float* sp = sum1 + c * F + f;
  float* qp = sumsq1 + c * F + f;
  atomicAdd(sp + 0, s0); atomicAdd(sp + 1, s1);
  atomicAdd(sp + 2, s2); atomicAdd(sp + 3, s3);
  atomicAdd(qp + 0, q0); atomicAdd(qp + 1, q1);
  atomicAdd(qp + 2, q2); atomicAdd(qp + 3, q3);
}

// ---------------------------------------------------------------------------
// Kernel B: fold BN1 + Linear1 -> a[c,f] = gamma1*rstd*W1,
//           t[c] = sum_f (beta1 - mean*rstd*gamma1)*W1 + b1[c]
// grid = C, block = 256 (4 f per thread)
// ---------------------------------------------------------------------------
__global__ __launch_bounds__(256) void k_fold1(const float* __restrict__ sum1,
                                               const float* __restrict__ sumsq1,
                                               const float* __restrict__ gamma1,
                                               const float* __restrict__ beta1,
                                               const float* __restrict__ W1,
                                               const float* __restrict__ b1,
                                               float* __restrict__ a,
                                               float* __restrict__ t) {
  const int c = blockIdx.x;
  const int f = threadIdx.x * 4;
  const float inv = 1.0f / (float)BS;

  float tl = 0.f;
#pragma unroll
  for (int k = 0; k < 4; ++k) {
    const int idx = c * F + f + k;
    const float m    = sum1[idx] * inv;
    const float var  = sumsq1[idx] * inv - m * m;
    const float rstd = rsqrtf(var + EPS);
    const float g    = gamma1[idx] * rstd;
    const float w    = W1[idx];
    a[idx] = g * w;
    tl += (beta1[idx] - m * g) * w;
  }
  __shared__ float red[256];
  red[threadIdx.x] = tl;
  __syncthreads();
  for (int s = 128; s > 0; s >>= 1) {
    if (threadIdx.x < s) red[threadIdx.x] += red[threadIdx.x + s];
    __syncthreads();
  }
  if (threadIdx.x == 0) t[c] = red[0] + b1[c];
}

// ---------------------------------------------------------------------------
// Kernel C: h[b,c] = sum_f x[b,c,f] * a[c,f] + t[c]
// grid = C * (BS/BTILE), block = 256 (8 waves, 16 b-rows per wave)
// Double-buffered async global->LDS staging + V_WMMA_F32_16X16X4_F32.
// WMMA mapping: A[16x4] = a-chunk replicated over rows, B[4x16] = x rows,
//   D[m,n] = h[b_n] for every m; accumulate over f; read acc[0], lanes 0..15.
// ---------------------------------------------------------------------------
__global__ __launch_bounds__(256) void k_gemv_wmma(const float* __restrict__ x,
                                                   const float* __restrict__ a,
                                                   const float* __restrict__ t,
                                                   float* __restrict__ h) {
  const int nb   = BS / BTILE;
  const int c    = blockIdx.x / nb;
  const int b0   = (blockIdx.x % nb) * BTILE;
  const int tid  = threadIdx.x;
  const int lane = tid & 31;
  const int wv   = tid >> 5;       // 0..7
  const int mrow = lane & 15;      // b-row within wave tile
  const int kh   = lane >> 4;      // K half: 0 -> K{0,1}, 1 -> K{2,3}

  __shared__ float a_lds[F];
  __shared__ float xbuf[2][BTILE * XROW];

  // stage a[c, 0:F] into LDS (4 floats/thread)
  {
    const float4 v = *(const float4*)(a + c * F + tid * 4);
    *(float4*)&a_lds[tid * 4] = v;
  }

  const float* gx = x + ((size_t)b0 * C + c) * F;

  auto stage = [&](int ch, int buf) {
    const float* gbase = gx + ch * FCHUNK;
#pragma unroll
    for (int i = 0; i < 8; ++i) {
      const int id  = tid + i * 256;     // 0..2047
      const int row = id >> 4;           // 0..127
      const int seg = id & 15;           // 16B segment within 64-float row
      const float* g = gbase + (size_t)row * (C * F) + seg * 4;
      float* l = &xbuf[buf][row * XROW + seg * 4];
#if HAVE_ASYNC
      __builtin_amdgcn_global_load_async_to_lds_b128(
          (AS_GLOBAL v4i*)(v4i*)(void*)g,   // drop const, reinterpret, addrspacecast
          (AS_LDS v4i*)(v4i*)(void*)l,
          0, 0);
#else
      *(float4*)l = *(const float4*)g;
#endif
    }
  };

  const float tc = t[c];

#if HAVE_WMMA
  v8f acc = {};
#else
  float accs = 0.f;
#endif

  stage(0, 0);
  for (int ch = 0; ch < NCHUNK; ++ch) {
    const int cur = ch & 1;
    if (ch + 1 < NCHUNK) stage(ch + 1, cur ^ 1);
#if HAVE_ASYNC
    if (ch + 1 < NCHUNK) { WAIT_ASYNC(8); } else { WAIT_ASYNC(0); }
#endif
    __syncthreads();   // staged chunk visible to all waves

    const float* xr = &xbuf[cur][(wv * 16 + mrow) * XROW];
    const float* ar = &a_lds[ch * FCHUNK];
#pragma unroll
    for (int f0 = 0; f0 < FCHUNK; f0 += 4) {
      const int ci = f0 + kh * 2;
      const v2f A = *(const v2f*)&ar[ci];   // a[f0+2kh], a[f0+2kh+1]
      const v2f B = *(const v2f*)&xr[ci];   // x[b_mrow, f0+2kh], +1
#if HAVE_WMMA
      acc = __builtin_amdgcn_wmma_f32_16x16x4_f32(
          /*neg_a=*/false, A, /*neg_b=*/false, B,
          /*c_mod=*/(short)0, acc, /*reuse_a=*/false, /*reuse_b=*/false);
#else
      accs += A.x * B.x + A.y * B.y;
#endif
    }
    __syncthreads();   // protect buffer before it is overwritten
  }

#if HAVE_WMMA
  const float hval = acc[0];                      // D[M=0, N=lane]
#else
  float hval = accs + __shfl_xor(accs, 16, 32);   // combine K halves
#endif
  if (lane < 16) {
    h[(size_t)(b0 + wv * 16 + lane) * C + c] = hval + tc;
  }
}

// ---------------------------------------------------------------------------
// Kernel D: BN2 stats per channel + fold with W2:
//   u[c] = rstd2*gamma2*W2, vterm[c] = (beta2 - mean2*rstd2*gamma2)*W2
// grid = C, block = 256
// ---------------------------------------------------------------------------
__global__ __launch_bounds__(256) void k_stats2(const float* __restrict__ h,
                                                const float* __restrict__ gamma2,
                                                const float* __restrict__ beta2,
                                                const float* __restrict__ W2,
                                                float* __restrict__ u,
                                                float* __restrict__ vterm) {
  const int c = blockIdx.x;
  float s = 0.f, q = 0.f;
  for (int b = threadIdx.x; b < BS; b += 256) {
    const float v = h[(size_t)b * C + c];
    s += v;
    q += v * v;
  }
  __shared__ float rs[256], rq[256];
  rs[threadIdx.x] = s;
  rq[threadIdx.x] = q;
  __syncthreads();
  for (int st = 128; st > 0; st >>= 1) {
    if (threadIdx.x < st) {
      rs[threadIdx.x] += rs[threadIdx.x + st];
      rq[threadIdx.x] += rq[threadIdx.x + st];
    }
    __syncthreads();
  }
  if (threadIdx.x == 0) {
    const float m    = rs[0] / (float)BS;
    const float var  = rq[0] / (float)BS - m * m;
    const float rstd = rsqrtf(var + EPS);
    const float g    = gamma2[c] * rstd;
    u[c]     = g * W2[c];
    vterm[c] = (beta2[c] - m * g) * W2[c];
  }
}

// ---------------------------------------------------------------------------
// Kernel E: out[b] = sum_c h[b,c]*u[c] + (sum_c vterm[c] + b2)
// grid = BS/256, block = 256
// ---------------------------------------------------------------------------
__global__ __launch_bounds__(256) void k_out(const float* __restrict__ h,
                                             const float* __restrict__ u,
                                             const float* __restrict__ vterm,
                                             const float* __restrict__ b2,
                                             float* __restrict__ out) {
  __shared__ float su[C];
  __shared__ float sv;
  if (threadIdx.x < C) su[threadIdx.x] = u[threadIdx.x];
  if (threadIdx.x == 0) {
    float v = b2[0];
    for (int cc = 0; cc < C; ++cc) v += vterm[cc];
    sv = v;
  }
  __syncthreads();
  const int b = blockIdx.x * 256 + threadIdx.x;
  float acc = sv;
  const float* hr = h + (size_t)b * C;
#pragma unroll
  for (int cc = 0; cc < C; ++cc) acc += hr[cc] * su[cc];
  out[b] = acc;
}

// ---------------------------------------------------------------------------
extern "C" void kernel_launch(void* const* d_in, const int* in_sizes, int n_in,
                              void* d_out, int out_size, void* d_ws, size_t ws_size,
                              hipStream_t stream) {
  (void)in_sizes; (void)n_in; (void)out_size; (void)ws_size;

  const float* x      = (const float*)d_in[0];
  const float* gamma1 = (const float*)d_in[1];
  const float* beta1  = (const float*)d_in[2];
  const float* W1     = (const float*)d_in[3];
  const float* b1     = (const float*)d_in[4];
  const float* gamma2 = (const float*)d_in[5];
  const float* beta2  = (const float*)d_in[6];
  const float* W2     = (const float*)d_in[7];
  const float* b2     = (const float*)d_in[8];

  float* ws     = (float*)d_ws;
  float* sum1   = ws + OFF_SUM;
  float* sumsq1 = ws + OFF_SUMSQ;
  float* a      = ws + OFF_A;
  float* t      = ws + OFF_T;
  float* u      = ws + OFF_U;
  float* vterm  = ws + OFF_V;
  float* h      = ws + OFF_H;

  // zero the atomic accumulators (graph-capturable)
  (void)hipMemsetAsync(sum1, 0, (size_t)2 * C * F * sizeof(float), stream);

  k_stats1<<<dim3(C * (BS / 128)), 256, 0, stream>>>(x, sum1, sumsq1);
  k_fold1<<<dim3(C), 256, 0, stream>>>(sum1, sumsq1, gamma1, beta1, W1, b1, a, t);
  k_gemv_wmma<<<dim3(C * (BS / BTILE)), 256, 0, stream>>>(x, a, t, h);
  k_stats2<<<dim3(C), 256, 0, stream>>>(h, gamma2, beta2, W2, u, vterm);
  k_out<<<dim3(BS / 256), 256, 0, stream>>>(h, u, vterm, b2, (float*)d_out);
}